// GeoFeatureExtractor_84490596647682
// MI455X (gfx1250) — compile-verified
//
#include <hip/hip_runtime.h>
#include <math.h>

#ifndef __has_builtin
#define __has_builtin(x) 0
#endif

#if __has_builtin(__builtin_amdgcn_global_load_async_to_lds_b32)
#define USE_ASYNC_LDS 1
typedef __attribute__((address_space(1))) int GInt;   // global-AS, per builtin signature
typedef __attribute__((address_space(3))) int LInt;   // LDS-AS
#else
#define USE_ASYNC_LDS 0
#endif

#if USE_ASYNC_LDS && __has_builtin(__builtin_amdgcn_global_load_async_to_lds_b128)
#define USE_ASYNC_B128 1
typedef int v4i __attribute__((ext_vector_type(4)));
typedef __attribute__((address_space(1))) v4i GV4;
typedef __attribute__((address_space(3))) v4i LV4;
#else
#define USE_ASYNC_B128 0
#endif

#define TT    64                 // time steps per block (64 threads = 2 wave32)
#define HALO  3                  // +-3 frames for cdiff(cdiff()) incl. edge padding
#define NS    (TT + 2*HALO)      // 70 LDS time slots
#define NJ    51
#define ROW   (NJ*3)             // 153 floats per slot (odd -> bank-conflict-free)
#define NF    114
#define TDIM  512

struct V3 { float x, y, z; };
__device__ __forceinline__ V3 sub3(V3 a, V3 b){ return {a.x-b.x, a.y-b.y, a.z-b.z}; }
__device__ __forceinline__ V3 add3(V3 a, V3 b){ return {a.x+b.x, a.y+b.y, a.z+b.z}; }
__device__ __forceinline__ V3 scl3(V3 a, float s){ return {a.x*s, a.y*s, a.z*s}; }
__device__ __forceinline__ float dot3(V3 a, V3 b){ return a.x*b.x + a.y*b.y + a.z*b.z; }
__device__ __forceinline__ V3 crs3(V3 a, V3 b){
  return {a.y*b.z - a.z*b.y, a.z*b.x - a.x*b.z, a.x*b.y - a.y*b.x};
}
__device__ __forceinline__ float nrm3(V3 a){ return sqrtf(dot3(a,a)); }          // _norm (no eps)
__device__ __forceinline__ float dst3(V3 a, V3 b){                               // _dist (eps in sqrt)
  V3 d = sub3(a,b); return sqrtf(dot3(d,d) + 1e-6f);
}
__device__ __forceinline__ float acosc(float c){                                 // _acos with clip
  c = fminf(fmaxf(c, -1.0f + 1e-6f), 1.0f - 1e-6f);
  return acosf(c);
}
__device__ __forceinline__ int clampi(int v, int lo, int hi){
  return v < lo ? lo : (v > hi ? hi : v);
}

__device__ __forceinline__ void cp32(const float* g, float* l) {
#if USE_ASYNC_LDS
  __builtin_amdgcn_global_load_async_to_lds_b32((GInt*)g, (LInt*)l, 0, 0);
#else
  *l = *g;
#endif
}
__device__ __forceinline__ void async_wait() {
#if USE_ASYNC_LDS
#if __has_builtin(__builtin_amdgcn_s_wait_asynccnt)
  __builtin_amdgcn_s_wait_asynccnt(0);
#else
  asm volatile("s_wait_asynccnt 0" ::: "memory");
#endif
#endif
}

__global__ __launch_bounds__(TT)
void GeoFeatureExtractor_84490596647682_kernel(const float* __restrict__ xyz,
                                               const float* __restrict__ fmask,
                                               const float* __restrict__ bmask,
                                               float* __restrict__ out)
{
  // +4 dwords: constant phase shift of 1 so global/LDS share mod-4 alignment for b128.
  __shared__ float L[NS * ROW + 4];
  float* Lp = L + 1;   // (gbase - dest) mod 4 == 1 for all tiles (bt0,t0 mult of 4; 459%4==3)

  const int tid       = threadIdx.x;
  const int tilesPerB = TDIM / TT;
  const int b         = blockIdx.x / tilesPerB;
  const int t0        = (blockIdx.x - b * tilesPerB) * TT;
  const size_t bt0    = (size_t)b * TDIM;

  // ---- Stage tile global -> LDS.
  // Contiguous slot span [sLo, sHi) copied flat with async b128; <=3 edge-clamped
  // halo rows (first/last tile only) replicated with dword copies.
  const int sLo = (t0 == 0) ? HALO : 0;
  const int sHi = (t0 + TT == TDIM) ? (NS - HALO) : NS;
  {
    const size_t gbase = (bt0 + (size_t)(t0 - HALO + sLo)) * ROW;  // dword index
    const float* g = xyz + gbase;
    float* d = Lp + sLo * ROW;
    const int nd = (sHi - sLo) * ROW;
    const int k  = (int)((4u - (unsigned)(gbase & 3u)) & 3u);      // head dwords to 16B
#if USE_ASYNC_B128
    const int m = (nd - k) >> 2;                                   // aligned 16B chunks
    for (int i = tid; i < m; i += TT)
      __builtin_amdgcn_global_load_async_to_lds_b128(
          (GV4*)(g + k + 4 * i), (LV4*)(d + k + 4 * i), 0, 0);
    for (int i = tid; i < k; i += TT)              cp32(g + i, d + i);
    for (int i = k + 4 * m + tid; i < nd; i += TT) cp32(g + i, d + i);
#else
    for (int i = tid; i < nd; i += TT)             cp32(g + i, d + i);
#endif
  }
  if (t0 == 0) {                       // replicate row gt=0 into slots 0..HALO-1
    const float* g = xyz + bt0 * ROW;
    for (int s = 0; s < HALO; ++s)
      for (int r = tid; r < ROW; r += TT) cp32(g + r, Lp + s * ROW + r);
  }
  if (t0 + TT == TDIM) {               // replicate row gt=T-1 into last HALO slots
    const float* g = xyz + (bt0 + TDIM - 1) * ROW;
    for (int s = NS - HALO; s < NS; ++s)
      for (int r = tid; r < ROW; r += TT) cp32(g + r, Lp + s * ROW + r);
  }
  async_wait();
  __syncthreads();

  // ---- Per-thread compute: one (b, t) sample from LDS.
  const int t  = t0 + tid;             // TDIM % TT == 0, always valid
  const int sC = HALO + tid;

  auto PS = [&](int s, int j) -> V3 {
    const float* p = &Lp[s * ROW + j * 3];
    return {p[0], p[1], p[2]};
  };
  auto P   = [&](int j) -> V3 { return PS(sC, j); };
  auto Pat = [&](int gt, int j) -> V3 { return PS(gt - t0 + HALO, j); };

  // vel(gt): central diff with _cdiff edge padding (vel[0]=vel[1], vel[T-1]=vel[T-2])
  auto VEL = [&](int j, int gt) -> V3 {
    const int c = clampi(gt, 1, TDIM - 2);
    return scl3(sub3(Pat(c + 1, j), Pat(c - 1, j)), 0.5f);
  };
  auto ACC = [&](int j, int gt) -> V3 {
    const int c = clampi(gt, 1, TDIM - 2);
    return scl3(sub3(VEL(j, c + 1), VEL(j, c - 1)), 0.5f);
  };

  const float fg = fmask[bt0 + t];
  const float bg = bmask[bt0 + t];

  float* o = out + (bt0 + (size_t)t) * NF;
  int f = 0;
  auto EMIT = [&](float v){ o[f++] = v; };

  // ---- tips / curls / cross / d_ti  (12 per hand, 24 total)
  for (int base = 0; base <= 21; base += 21) {
    V3 tT = P(base + 4),  iT = P(base + 8),  mT = P(base + 12);
    V3 rT = P(base + 16), pT = P(base + 20);
    EMIT(dst3(tT, iT)); EMIT(dst3(iT, mT)); EMIT(dst3(mT, rT));
    EMIT(dst3(rT, pT)); EMIT(dst3(tT, pT));
    EMIT(dst3(P(base + 2),  tT) / (dst3(P(base + 2),  P(base + 3))  + 1e-4f));
    EMIT(dst3(P(base + 5),  iT) / (dst3(P(base + 5),  P(base + 6))  + 1e-4f));
    EMIT(dst3(P(base + 9),  mT) / (dst3(P(base + 9),  P(base + 10)) + 1e-4f));
    EMIT(dst3(P(base + 13), rT) / (dst3(P(base + 13), P(base + 14)) + 1e-4f));
    EMIT(dst3(P(base + 17), pT) / (dst3(P(base + 17), P(base + 18)) + 1e-4f));
    EMIT(iT.x - mT.x);
    EMIT(dst3(tT, P(base + 5)));
  }

  // ---- face distances * fg  (10)
  V3 w0 = P(0), w1 = P(21), nose = P(42), chin = P(43), fh = P(44);
  {
    V3 it0 = P(8), it1 = P(29);
    EMIT(dst3(w0, nose) * fg); EMIT(dst3(w0, chin) * fg); EMIT(dst3(w0, fh) * fg);
    EMIT(dst3(w1, nose) * fg); EMIT(dst3(w1, chin) * fg); EMIT(dst3(w1, fh) * fg);
    EMIT(dst3(it0, nose) * fg); EMIT(dst3(it0, fh) * fg);
    EMIT(dst3(it1, nose) * fg); EMIT(dst3(it1, fh) * fg);
  }

  // ---- joint triplet angles (15 per hand, 30)
  {
    const int TP[15] = {0,1,2,0,5,6,0,9,10,0,13,14,0,17,18};
    const int TJ[15] = {1,2,3,5,6,7,9,10,11,13,14,15,17,18,19};
    const int TC[15] = {2,3,4,6,7,8,10,11,12,14,15,16,18,19,20};
    for (int base = 0; base <= 21; base += 21) {
      for (int k = 0; k < 15; ++k) {
        V3 vj = P(base + TJ[k]);
        V3 v1 = sub3(P(base + TP[k]), vj);
        V3 v2 = sub3(P(base + TC[k]), vj);
        EMIT(acosc(dot3(v1, v2) / (nrm3(v1) * nrm3(v2) + 1e-6f)));
      }
    }
  }

  // ---- palm normals (3 per hand, 6)
  V3 n0, n1;
  {
    V3 c0 = crs3(sub3(P(5),  w0), sub3(P(17), w0));
    n0 = scl3(c0, 1.0f / (nrm3(c0) + 1e-6f));
    EMIT(n0.x); EMIT(n0.y); EMIT(n0.z);
    V3 c1 = crs3(sub3(P(26), w1), sub3(P(38), w1));
    n1 = scl3(c1, 1.0f / (nrm3(c1) + 1e-6f));
    EMIT(n1.x); EMIT(n1.y); EMIT(n1.z);
  }

  // ---- MCP spread angles (3 per hand, 6)
  {
    const int S[4] = {5, 9, 13, 17};
    for (int base = 0; base <= 21; base += 21) {
      V3 w = P(base);
      for (int i = 0; i < 3; ++i) {
        V3 v1 = sub3(P(base + S[i]),     w);
        V3 v2 = sub3(P(base + S[i + 1]), w);
        EMIT(acosc(dot3(v1, v2) / (nrm3(v1) * nrm3(v2) + 1e-6f)));
      }
    }
  }

  // ---- normal . up / fwd  (4)
  EMIT(n0.y); EMIT(n0.z); EMIT(n1.y); EMIT(n1.z);

  // ---- velocity direction (3 per hand, 6); note: maximum(|v|,1e-6)
  V3 v0 = VEL(0, t), v1 = VEL(21, t);
  {
    float m0 = fmaxf(nrm3(v0), 1e-6f);
    EMIT(v0.x / m0); EMIT(v0.y / m0); EMIT(v0.z / m0);
    float m1 = fmaxf(nrm3(v1), 1e-6f);
    EMIT(v1.x / m1); EMIT(v1.y / m1); EMIT(v1.z / m1);
  }

  // ---- velocity turn angle (2); zero-padded at t = T-1
  {
    V3 a0 = scl3(v0, 1.0f / (nrm3(v0) + 1e-6f));
    V3 b0 = VEL(0, t + 1);  b0 = scl3(b0, 1.0f / (nrm3(b0) + 1e-6f));
    EMIT(t < TDIM - 1 ? acosc(dot3(a0, b0)) : 0.0f);
    V3 a1 = scl3(v1, 1.0f / (nrm3(v1) + 1e-6f));
    V3 b1 = VEL(21, t + 1); b1 = scl3(b1, 1.0f / (nrm3(b1) + 1e-6f));
    EMIT(t < TDIM - 1 ? acosc(dot3(a1, b1)) : 0.0f);
  }

  // ---- inter-wrist distance + relative direction (3)
  EMIT(dst3(w0, w1));
  {
    V3 rel = sub3(w1, w0);
    float rm = nrm3(rel) + 1e-6f;
    EMIT(rel.x / rm); EMIT(rel.y / rm);
  }

  // ---- cdiff of wrist-to-nose distance (2)
  {
    const int c = clampi(t, 1, TDIM - 2);
    for (int base = 0; base <= 21; base += 21) {
      float gp = nrm3(sub3(Pat(c + 1, base), Pat(c + 1, 42)));
      float gm = nrm3(sub3(Pat(c - 1, base), Pat(c - 1, 42)));
      EMIT((gp - gm) * 0.5f);
    }
  }

  // ---- fingertip x-splays (4)
  EMIT(P(8).x  - P(12).x); EMIT(P(12).x - P(16).x);
  EMIT(P(29).x - P(33).x); EMIT(P(33).x - P(37).x);

  // ---- hand-velocity alignment (1)
  {
    V3 ld = scl3(v0, 1.0f / (nrm3(v0) + 1e-6f));
    V3 rd = scl3(v1, 1.0f / (nrm3(v1) + 1e-6f));
    EMIT(dot3(ld, rd));
  }

  // ---- speeds (2), accel magnitudes (2)
  EMIT(nrm3(v0)); EMIT(nrm3(v1));
  EMIT(nrm3(ACC(0, t))); EMIT(nrm3(ACC(21, t)));

  // ---- proximity sigmoid (1)
  {
    float hd = nrm3(sub3(w0, w1));
    float z  = 5.0f * (0.05f - hd);
    EMIT(1.0f / (1.0f + expf(-z)));
  }

  // ---- body features * bg (11)
  {
    V3 lsh = P(45), rsh = P(46);
    V3 shm = scl3(add3(lsh, rsh), 0.5f);
    float shw = dst3(lsh, rsh);
    float inv = 1.0f / (shw + 1e-6f);
    EMIT(bg * (w0.y - shm.y) * inv);
    EMIT(bg * (w0.x - shm.x) * inv);
    EMIT(bg * dst3(w0, lsh));
    EMIT(bg * dst3(w0, P(47)));
    EMIT(bg * (w1.y - shm.y) * inv);
    EMIT(bg * (w1.x - shm.x) * inv);
    EMIT(bg * dst3(w1, rsh));
    EMIT(bg * dst3(w1, P(48)));
    EMIT(bg * shw);
    V3 mouth = scl3(add3(P(49), P(50)), 0.5f);
    EMIT(bg * dst3(w0, mouth));
    EMIT(bg * dst3(w1, mouth));
  }
  // f == 114 here
}

extern "C" void kernel_launch(void* const* d_in, const int* in_sizes, int n_in,
                              void* d_out, int out_size, void* d_ws, size_t ws_size,
                              hipStream_t stream) {
  const float* xyz   = (const float*)d_in[0];
  const float* fmask = (const float*)d_in[1];
  const float* bmask = (const float*)d_in[2];
  float* out = (float*)d_out;

  int B = in_sizes[0] / (TDIM * NJ * 3);
  if (B < 1) B = 1;
  const int nBlocks = B * (TDIM / TT);

  hipLaunchKernelGGL(GeoFeatureExtractor_84490596647682_kernel,
                     dim3(nBlocks), dim3(TT), 0, stream,
                     xyz, fmask, bmask, out);
}